// NCCForces3d_56745107915362
// MI455X (gfx1250) — compile-verified
//
#include <hip/hip_runtime.h>

// ---------------------------------------------------------------------------
// Fused NCC-forces 3D for MI455X (gfx1250, wave32).
//   grid: (16,16,32) blocks of 256 threads; each block produces a 16x16x8 tile.
//   Staging: global->LDS async (ASYNCcnt) of m/f halo tiles (10x18x18 each).
//   Box sums separable: x in VALU (pass 1, producing the 5 product streams),
//   y via V_WMMA_F32_16X16X4_F32 band-matrix GEMM (pass 2), z as 3-plane add.
//   Output streamed with non-temporal stores (inputs, 134 MB, stay in 192MB L2).
//   LDS: sm/sf 25.9KB + sx 64KB (rows padded 18->20 so B-loads are unguarded)
//        + sy 52.2KB  =~ 142 KB  -> 2 blocks per 320 KB WGP.
// ---------------------------------------------------------------------------

#define W 256
#define TZ 8
#define RZ (TZ + 2)             // 10 z-planes incl halo
#define RY 18
#define RYP 20                  // padded sx rows: K=20 = 5 WMMA chunks, no guards
#define RX 18
#define RTOT (RZ * RY * RX)     // 3240 staged elements per image
#define QS (RZ * RYP * 16)      // 3200: per-quantity sx stride
#define SYQ (RZ * 16 * 16)      // 2560: per-quantity sy stride
#define NTILE (5 * RZ)          // 50 real WMMA tiles
#define EPSF 1e-6f

typedef __attribute__((ext_vector_type(2))) float v2f;
typedef __attribute__((ext_vector_type(8))) float v8f;

#if defined(__gfx1250__) && __has_builtin(__builtin_amdgcn_global_load_async_to_lds_b32)
#define USE_ASYNC 1
#endif
#if defined(__gfx1250__) && __has_builtin(__builtin_amdgcn_wmma_f32_16x16x4_f32)
#define USE_WMMA 1
#endif

typedef __attribute__((address_space(1))) int as1_int;   // global
typedef __attribute__((address_space(3))) int as3_int;   // LDS

__device__ __forceinline__ void stage_one(const float* g, float* l) {
#ifdef USE_ASYNC
  // gfx1250 async VMEM->LDS path (tracked by ASYNCcnt), per-lane addresses.
  __builtin_amdgcn_global_load_async_to_lds_b32(
      (as1_int*)g, (as3_int*)l, 0, 0);
#else
  *l = *g;
#endif
}

__device__ __forceinline__ void wait_stage() {
#ifdef USE_ASYNC
#if __has_builtin(__builtin_amdgcn_s_wait_asynccnt)
  __builtin_amdgcn_s_wait_asynccnt(0);
#else
  asm volatile("s_wait_asynccnt 0" ::: "memory");
#endif
#endif
}

// torch/jnp.gradient semantics: central interior, one-sided at the ends.
__device__ __forceinline__ float grad1d(float xm, float xc, float xp, int g) {
  if (g == 0) return xp - xc;
  if (g == W - 1) return xc - xm;
  return 0.5f * (xp - xm);
}

__global__ __launch_bounds__(256) void ncc_forces_fused(
    const float* __restrict__ mov, const float* __restrict__ fix,
    const unsigned char* __restrict__ mmask,   // jnp bool -> 1 byte/elem
    const unsigned char* __restrict__ fmask,
    float* __restrict__ out) {
  __shared__ float sm[RTOT];
  __shared__ float sf[RTOT];
  __shared__ float sx[5 * QS];
  __shared__ float sy[5 * SYQ + 256];   // +256: scratch slot for padded tiles

  const int tid = threadIdx.x;
  const int x0 = blockIdx.x * 16;
  const int y0 = blockIdx.y * 16;
  const int z0 = blockIdx.z * TZ;

  // ---- stage m,f halo tile; zero-fill first only for boundary blocks ------
  const bool edge = (blockIdx.x == 0) | (blockIdx.y == 0) | (blockIdx.z == 0) |
                    (blockIdx.x == (W / 16 - 1)) | (blockIdx.y == (W / 16 - 1)) |
                    (blockIdx.z == (W / TZ - 1));
  if (edge) {
    for (int i = tid; i < RTOT; i += 256) { sm[i] = 0.f; sf[i] = 0.f; }
    __syncthreads();
  }
  for (int i = tid; i < RTOT; i += 256) {
    const int lz = i / (RY * RX);
    const int r  = i - lz * (RY * RX);
    const int ly = r / RX;
    const int lx = r - ly * RX;
    const int gz = z0 - 1 + lz, gy = y0 - 1 + ly, gx = x0 - 1 + lx;
    if (((unsigned)gz < W) & ((unsigned)gy < W) & ((unsigned)gx < W)) {
      const int gidx = (gz * W + gy) * W + gx;
      stage_one(mov + gidx, sm + i);
      stage_one(fix + gidx, sf + i);
    }
  }
  wait_stage();
  __syncthreads();

  // ---- pass 1: x-direction 3-tap sums of the 5 product streams ------------
  // Layout sx[q][lz][ly(0..19)][lx]; rows 18,19 are zero pad for the GEMM.
  for (int i = tid; i < RZ * RYP * 16; i += 256) {
    const int lz = i / (RYP * 16);
    const int r  = i - lz * (RYP * 16);
    const int ly = r >> 4;
    const int lx = r & 15;
    if (ly < RY) {
      const int b = (lz * RY + ly) * RX + lx;  // reads lx, lx+1, lx+2 (halo)
      const float m0 = sm[b], m1 = sm[b + 1], m2 = sm[b + 2];
      const float f0 = sf[b], f1 = sf[b + 1], f2 = sf[b + 2];
      sx[0 * QS + i] = m0 + m1 + m2;
      sx[1 * QS + i] = f0 + f1 + f2;
      sx[2 * QS + i] = m0 * m0 + m1 * m1 + m2 * m2;
      sx[3 * QS + i] = f0 * f0 + f1 * f1 + f2 * f2;
      sx[4 * QS + i] = m0 * f0 + m1 * f1 + m2 * f2;
    } else {
      sx[0 * QS + i] = 0.f; sx[1 * QS + i] = 0.f; sx[2 * QS + i] = 0.f;
      sx[3 * QS + i] = 0.f; sx[4 * QS + i] = 0.f;
    }
  }
  __syncthreads();

  // ---- pass 2: y-direction sum as 16x16 = T(16x20) * SX(20x16) ------------
#ifdef USE_WMMA
  {
    const int wave = tid >> 5;
    const int lane = tid & 31;
    const int n  = lane & 15;   // B column == D column == A row (M) for this lane
    const int hi = lane >> 4;   // ISA A layout: lanes 0-15 hold K0/K1, 16-31 hold K2/K3
    // Hoist the band matrix: A depends only on (lane, chunk), not the tile.
    // T[m][k] = 1 iff m <= k <= m+2  (sy[y] = sum sx[y..y+2]); rows 18,19 = 0 pad.
    v2f Aband[5];
#pragma unroll
    for (int c = 0; c < 5; ++c) {
      const int k0 = 4 * c + 2 * hi;
      const int k1 = k0 + 1;
      Aband[c].x = (k0 >= n && k0 <= n + 2) ? 1.f : 0.f;
      Aband[c].y = (k1 >= n && k1 <= n + 2) ? 1.f : 0.f;
    }
    // Uniform trip count: 56 padded tiles / 8 waves = 7 iterations, EXEC all-1s
    // throughout; tiles >= NTILE read tile 49 (harmless) and write the pad slot.
    for (int it = 0; it < 7; ++it) {
      const int t  = wave + 8 * it;
      const int tc = (t < NTILE) ? t : (NTILE - 1);
      const int q  = tc / RZ;
      const int lz = tc - q * RZ;
      const float* sxq = sx + q * QS + lz * (RYP * 16);
      v8f acc = {0.f, 0.f, 0.f, 0.f, 0.f, 0.f, 0.f, 0.f};
#pragma unroll
      for (int c = 0; c < 5; ++c) {                   // K = 20, no row guards
        const int k0 = 4 * c + 2 * hi;
        v2f B;
        B.x = sxq[k0 * 16 + n];
        B.y = sxq[k0 * 16 + 16 + n];
        acc = __builtin_amdgcn_wmma_f32_16x16x4_f32(false, Aband[c], false, B,
                                                    (short)0, acc, false, false);
      }
      // D layout: VGPR v holds row m = v (lanes 0-15) / m = v+8 (lanes 16-31)
      float* syq = (t < NTILE) ? (sy + q * SYQ + lz * 256) : (sy + 5 * SYQ);
#pragma unroll
      for (int v = 0; v < 8; ++v) syq[(v + 8 * hi) * 16 + n] = acc[v];
    }
  }
#else
  for (int i = tid; i < RZ * 256; i += 256) {
    const int lz = i >> 8;
    const int yx = i & 255;
    const int y = yx >> 4, x = yx & 15;
#pragma unroll
    for (int q = 0; q < 5; ++q) {
      const float* p = sx + q * QS + (lz * RYP + y) * 16 + x;
      sy[q * SYQ + i] = p[0] + p[16] + p[32];
    }
  }
#endif
  __syncthreads();

  // ---- pass 3: z-sum + statistics + gradients + force, streamed out -------
  const int ox = tid & 15;
  const int oy = tid >> 4;
  const int NVOX = W * W * W;
  for (int zz = 0; zz < TZ; ++zz) {
    float S[5];
#pragma unroll
    for (int q = 0; q < 5; ++q) {
      const float* p = sy + q * SYQ + zz * 256 + oy * 16 + ox;
      S[q] = p[0] + p[256] + p[512];
    }
    const float sum_m = S[0], sum_f = S[1];
    const float sum_mm = S[2], sum_ff = S[3], sum_mf = S[4];
    const float inv27 = 1.0f / 27.0f;
    const float mean_m = sum_m * inv27;
    const float mean_f = sum_f * inv27;
    const float var_m = sum_mm - 2.f * mean_m * sum_m + 27.f * mean_m * mean_m;
    const float var_f = sum_ff - 2.f * mean_f * sum_f + 27.f * mean_f * mean_f;
    const float cross = sum_mf - mean_f * sum_m - mean_m * sum_f +
                        27.f * mean_m * mean_f;

    const int ci = ((zz + 1) * RY + (oy + 1)) * RX + (ox + 1);
    const float mc = sm[ci], fc = sf[ci];
    const int gz = z0 + zz, gy = y0 + oy, gx = x0 + ox;

    const float gmz = grad1d(sm[ci - RY * RX], mc, sm[ci + RY * RX], gz);
    const float gmy = grad1d(sm[ci - RX],      mc, sm[ci + RX],      gy);
    const float gmx = grad1d(sm[ci - 1],       mc, sm[ci + 1],       gx);
    const float gfz = grad1d(sf[ci - RY * RX], fc, sf[ci + RY * RX], gz);
    const float gfy = grad1d(sf[ci - RX],      fc, sf[ci + RX],      gy);
    const float gfx = grad1d(sf[ci - 1],       fc, sf[ci + 1],       gx);

    const int gidx = (gz * W + gy) * W + gx;
    const float wm = mmask[gidx] ? 1.f : 0.f;
    const float wf = fmask[gidx] ? 1.f : 0.f;

    const float tgz = 0.5f * (gmz * wm + gfz * wf);
    const float tgy = 0.5f * (gmy * wm + gfy * wf);
    const float tgx = 0.5f * (gmx * wm + gfx * wf);

    const float factor = 2.f * cross / (var_m * var_f + EPSF) *
                         ((mc - mean_m) - cross / (var_f * (fc - mean_f) + EPSF));

    // NT stores: don't evict the two input volumes (134 MB) from the 192 MB L2.
    __builtin_nontemporal_store(-factor * tgz, out + gidx);
    __builtin_nontemporal_store(-factor * tgy, out + NVOX + gidx);
    __builtin_nontemporal_store(-factor * tgx, out + 2 * NVOX + gidx);
  }
}

extern "C" void kernel_launch(void* const* d_in, const int* in_sizes, int n_in,
                              void* d_out, int out_size, void* d_ws, size_t ws_size,
                              hipStream_t stream) {
  const float* mov = (const float*)d_in[0];
  const float* fix = (const float*)d_in[1];
  const unsigned char* mmask = (const unsigned char*)d_in[2];
  const unsigned char* fmask = (const unsigned char*)d_in[3];
  float* out = (float*)d_out;
  dim3 grid(W / 16, W / 16, W / TZ);   // (16,16,32)
  ncc_forces_fused<<<grid, dim3(256), 0, stream>>>(mov, fix, mmask, fmask, out);
  (void)in_sizes; (void)n_in; (void)out_size; (void)d_ws; (void)ws_size;
}